// ProbAttention_17910013624419
// MI455X (gfx1250) — compile-verified
//
#include <hip/hip_runtime.h>
#include <math.h>
#include <stdint.h>

// Problem constants (from reference): B=4, L=2048, H=8, D=64, SAMPLE_K=N_TOP=40
#define BSZ   4
#define LSEQ  2048
#define HHEAD 8
#define DDIM  64
#define NSAMP 40
#define NTOP  40
#define MROWS 48      // NTOP padded to 3 tiles of 16
#define CHUNK 64      // keys per flash chunk
#define NCHUNK (LSEQ / CHUNK)
#define DPAD  68      // LDS pitch (floats): multiple of 4, conflict-free strided reads

typedef float v2f __attribute__((ext_vector_type(2)));
typedef float v8f __attribute__((ext_vector_type(8)));

// Async global->LDS copy of 16 bytes per lane (gfx1250 ASYNCcnt path).
// vdst = 32-bit LDS byte offset (low 32 bits of the flat LDS address),
// vaddr = 64-bit global address.
__device__ __forceinline__ void async_cp16(const float* gptr, const float* lptr)
{
    const unsigned lds_off = (unsigned)(uintptr_t)lptr;
    asm volatile("global_load_async_to_lds_b128 %0, %1, off"
                 :: "v"(lds_off), "v"(gptr) : "memory");
}

// ---------------------------------------------------------------------------
// Kernel 1: sparsity measure M[bh][l] = max_s(q.k_s) - sum_s(q.k_s)/L
// One wave32 per query; lanes split D=64 (2 elems each).
// ---------------------------------------------------------------------------
__global__ __launch_bounds__(256) void compute_M_kernel(
    const float* __restrict__ Q, const float* __restrict__ K,
    const int* __restrict__ samp, float* __restrict__ M)
{
    const int lane = threadIdx.x & 31;
    const int wv   = threadIdx.x >> 5;
    const int gq   = blockIdx.x * 8 + wv;          // 0 .. B*H*L-1  (bh*L + l)
    const int bh   = gq >> 11;                     // /L
    const int l    = gq & (LSEQ - 1);
    const int b    = bh >> 3;
    const int h    = bh & 7;

    const float* qp = Q + (((size_t)b * LSEQ + l) * HHEAD + h) * DDIM;
    const float q0 = qp[lane];
    const float q1 = qp[lane + 32];

    float mx = -INFINITY, sm = 0.0f;
    for (int s = 0; s < NSAMP; ++s) {
        const int ki = samp[l * NSAMP + s];
        const float* kp = K + (((size_t)b * LSEQ + ki) * HHEAD + h) * DDIM;
        float pa = q0 * kp[lane] + q1 * kp[lane + 32];
        #pragma unroll
        for (int off = 16; off > 0; off >>= 1)
            pa += __shfl_xor(pa, off);
        mx = fmaxf(mx, pa);
        sm += pa;
    }
    if (lane == 0)
        M[gq] = mx - sm * (1.0f / (float)LSEQ);
}

// ---------------------------------------------------------------------------
// Kernel 2: top-40 indices per (b,h) via 40 argmax passes over LDS copy of M.
// ---------------------------------------------------------------------------
__global__ __launch_bounds__(256) void topk_kernel(
    const float* __restrict__ M, int* __restrict__ Mtop)
{
    __shared__ float vals[LSEQ];
    __shared__ float rv[256];
    __shared__ int   ri[256];
    const int bh = blockIdx.x;
    const int t  = threadIdx.x;

    for (int j = t; j < LSEQ; j += 256) vals[j] = M[(size_t)bh * LSEQ + j];
    __syncthreads();

    for (int it = 0; it < NTOP; ++it) {
        float bv = -INFINITY; int bi = 0;
        for (int j = t; j < LSEQ; j += 256) {
            float v = vals[j];
            if (v > bv) { bv = v; bi = j; }
        }
        rv[t] = bv; ri[t] = bi;
        __syncthreads();
        for (int off = 128; off > 0; off >>= 1) {
            if (t < off && rv[t + off] > rv[t]) { rv[t] = rv[t + off]; ri[t] = ri[t + off]; }
            __syncthreads();
        }
        if (t == 0) { Mtop[bh * NTOP + it] = ri[0]; vals[ri[0]] = -INFINITY; }
        __syncthreads();
    }
}

// ---------------------------------------------------------------------------
// Kernel 3a: V column mean per (b,h).  Kernel 3b: broadcast mean to output.
// ---------------------------------------------------------------------------
__global__ __launch_bounds__(64) void vmean_kernel(
    const float* __restrict__ V, float* __restrict__ vmean)
{
    const int bh = blockIdx.x, d = threadIdx.x;
    const int b = bh >> 3, h = bh & 7;
    const float* p = V + ((size_t)b * LSEQ * HHEAD + h) * DDIM + d;
    float s = 0.0f;
    for (int l = 0; l < LSEQ; ++l) s += p[(size_t)l * (HHEAD * DDIM)];
    vmean[bh * DDIM + d] = s * (1.0f / (float)LSEQ);
}

__global__ __launch_bounds__(256) void bcast_kernel(
    const float* __restrict__ vmean, float* __restrict__ out)
{
    const size_t i = (size_t)blockIdx.x * 256 + threadIdx.x; // B*L*H*D = 4M elems
    const int d = (int)(i & 63);
    const int h = (int)((i >> 6) & 7);
    const int b = (int)(i >> 20);                            // L*H*D = 2^20
    out[i] = vmean[(b * HHEAD + h) * DDIM + d];
}

// ---------------------------------------------------------------------------
// Kernel 4: flash-style attention for the top-40 queries of each (b,h).
// V_WMMA_F32_16X16X4_F32 for Q.K^T and P.V; K/V chunks double-buffered in LDS
// and prefetched with GLOBAL_LOAD_ASYNC_TO_LDS_B128 (ASYNCcnt pipeline).
// 4 waves/block, each wave owns a 16-wide column slice.
// ---------------------------------------------------------------------------
__global__ __launch_bounds__(128) void attn_topk_kernel(
    const float* __restrict__ Q, const float* __restrict__ K,
    const float* __restrict__ V, const int* __restrict__ Mtop,
    float* __restrict__ out)
{
    __shared__ __attribute__((aligned(16))) float Qs[MROWS][DPAD];
    __shared__ __attribute__((aligned(16))) float Ks[2][CHUNK][DPAD];
    __shared__ __attribute__((aligned(16))) float Vs[2][CHUNK][DPAD];
    __shared__ __attribute__((aligned(16))) float Ss[MROWS][DPAD];
    __shared__ float rm[MROWS], rl[MROWS], ral[MROWS];

    const int bh   = blockIdx.x;
    const int b    = bh >> 3, h = bh & 7;
    const int tid  = threadIdx.x;
    const int wv   = tid >> 5;         // wave id: column slice [16*wv, 16*wv+16)
    const int lane = tid & 31;
    const int col  = lane & 15;        // N (or M-row for A operand)
    const int kh   = (lane >> 4) << 1; // K-half offset: 0 or 2
    const int hi8  = (lane >> 4) << 3; // C/D row offset: 0 or 8

    // Async prefetch of one K/V chunk into LDS buffer bf (16 issues per wave).
    auto issue_chunk = [&](int ck, int bf) {
        const float* Kb = K + (((size_t)b * LSEQ + ck * CHUNK) * HHEAD + h) * DDIM;
        const float* Vb = V + (((size_t)b * LSEQ + ck * CHUNK) * HHEAD + h) * DDIM;
        #pragma unroll
        for (int u = tid; u < CHUNK * 16; u += 128) {
            const int r = u >> 4, c4 = (u & 15) << 2;
            async_cp16(Kb + (size_t)r * (HHEAD * DDIM) + c4, &Ks[bf][r][c4]);
        }
        #pragma unroll
        for (int u = tid; u < CHUNK * 16; u += 128) {
            const int r = u >> 4, c4 = (u & 15) << 2;
            async_cp16(Vb + (size_t)r * (HHEAD * DDIM) + c4, &Vs[bf][r][c4]);
        }
    };

    // Gather Q_reduce rows (pre-scaled by 1/sqrt(D)), zero-pad rows 40..47.
    for (int u = tid; u < MROWS * 16; u += 128) {
        const int r = u >> 4, c4 = (u & 15) << 2;
        float4 v = make_float4(0.f, 0.f, 0.f, 0.f);
        if (r < NTOP) {
            const int li = Mtop[bh * NTOP + r];
            v = *(const float4*)(Q + (((size_t)b * LSEQ + li) * HHEAD + h) * DDIM + c4);
        }
        Qs[r][c4 + 0] = v.x * 0.125f; Qs[r][c4 + 1] = v.y * 0.125f;
        Qs[r][c4 + 2] = v.z * 0.125f; Qs[r][c4 + 3] = v.w * 0.125f;
    }
    if (tid < MROWS) { rm[tid] = -INFINITY; rl[tid] = 0.0f; }

    v8f o[3];
    #pragma unroll
    for (int mt = 0; mt < 3; ++mt) o[mt] = (v8f){};

    issue_chunk(0, 0);                       // prologue prefetch

    for (int ck = 0; ck < NCHUNK; ++ck) {
        const int bf = ck & 1;
        if (ck + 1 < NCHUNK) {
            issue_chunk(ck + 1, bf ^ 1);     // buf^1 free: trailing barrier of prev iter
            // 32 outstanding per wave -> wait until only next chunk's 16 remain
            asm volatile("s_wait_asynccnt 0x10" ::: "memory");
        } else {
            asm volatile("s_wait_asynccnt 0x0" ::: "memory");
        }
        __syncthreads();                     // all waves' shares of chunk ck resident

        // S(48 x 64) = Qs . Ks^T  — each wave computes its 16-key slice.
        #pragma unroll
        for (int mt = 0; mt < 3; ++mt) {
            v8f c = (v8f){};
            #pragma unroll
            for (int k0 = 0; k0 < DDIM; k0 += 4) {
                v2f a, bb;
                a.x  = Qs[mt * 16 + col][k0 + kh];
                a.y  = Qs[mt * 16 + col][k0 + kh + 1];
                bb.x = Ks[bf][wv * 16 + col][k0 + kh];      // B[k][n] = K^T
                bb.y = Ks[bf][wv * 16 + col][k0 + kh + 1];
                c = __builtin_amdgcn_wmma_f32_16x16x4_f32(
                        false, a, false, bb, (short)0, c, false, false);
            }
            #pragma unroll
            for (int i = 0; i < 8; ++i)
                Ss[mt * 16 + i + hi8][wv * 16 + col] = c[i];
        }
        __syncthreads();

        // Online softmax update: one thread per row.
        if (tid < MROWS) {
            const float mold = rm[tid];
            float mx = mold;
            #pragma unroll 8
            for (int j = 0; j < CHUNK; ++j) mx = fmaxf(mx, Ss[tid][j]);
            const float alpha = __expf(mold - mx);
            float s = 0.0f;
            #pragma unroll 8
            for (int j = 0; j < CHUNK; ++j) {
                const float p = __expf(Ss[tid][j] - mx);
                Ss[tid][j] = p;
                s += p;
            }
            rl[tid] = rl[tid] * alpha + s;
            rm[tid] = mx;
            ral[tid] = alpha;
        }
        __syncthreads();

        // O = O*alpha + P . V  — each wave accumulates its 16-wide d slice.
        #pragma unroll
        for (int mt = 0; mt < 3; ++mt) {
            v8f c = o[mt];
            #pragma unroll
            for (int i = 0; i < 8; ++i) c[i] *= ral[mt * 16 + i + hi8];
            #pragma unroll
            for (int k0 = 0; k0 < CHUNK; k0 += 4) {
                v2f a, bb;
                a.x  = Ss[mt * 16 + col][k0 + kh];
                a.y  = Ss[mt * 16 + col][k0 + kh + 1];
                bb.x = Vs[bf][k0 + kh][wv * 16 + col];      // B[k][n] = V
                bb.y = Vs[bf][k0 + kh + 1][wv * 16 + col];
                c = __builtin_amdgcn_wmma_f32_16x16x4_f32(
                        false, a, false, bb, (short)0, c, false, false);
            }
            o[mt] = c;
        }
        __syncthreads();   // all reads of buf/Ss done before next prefetch/overwrite
    }

    // Normalize and scatter the 40 updated rows into the broadcast output.
    #pragma unroll
    for (int mt = 0; mt < 3; ++mt) {
        #pragma unroll
        for (int i = 0; i < 8; ++i) {
            const int m = mt * 16 + i + hi8;
            if (m < NTOP) {
                const int li = Mtop[bh * NTOP + m];
                out[(((size_t)b * LSEQ + li) * HHEAD + h) * DDIM + wv * 16 + col] =
                    o[mt][i] / rl[m];
            }
        }
    }
}

// ---------------------------------------------------------------------------
extern "C" void kernel_launch(void* const* d_in, const int* in_sizes, int n_in,
                              void* d_out, int out_size, void* d_ws, size_t ws_size,
                              hipStream_t stream)
{
    (void)in_sizes; (void)n_in; (void)out_size; (void)ws_size;
    const float* Q    = (const float*)d_in[0];
    const float* K    = (const float*)d_in[1];
    const float* V    = (const float*)d_in[2];
    const int*   samp = (const int*)d_in[3];
    float* out = (float*)d_out;

    // workspace layout (~275 KB)
    float* M     = (float*)d_ws;                                        // B*H*L floats
    int*   Mtop  = (int*)((char*)d_ws + (size_t)BSZ * HHEAD * LSEQ * 4);// B*H*NTOP ints
    float* vmean = (float*)((char*)Mtop + (size_t)BSZ * HHEAD * NTOP * 4);

    compute_M_kernel<<<(BSZ * HHEAD * LSEQ) / 8, 256, 0, stream>>>(Q, K, samp, M);
    topk_kernel<<<BSZ * HHEAD, 256, 0, stream>>>(M, Mtop);
    vmean_kernel<<<BSZ * HHEAD, 64, 0, stream>>>(V, vmean);
    bcast_kernel<<<(BSZ * LSEQ * HHEAD * DDIM) / 256, 256, 0, stream>>>(vmean, out);
    attn_topk_kernel<<<BSZ * HHEAD, 128, 0, stream>>>(Q, K, V, Mtop, out);
}